// WassersteinSimilarity_16415365005840
// MI455X (gfx1250) — compile-verified
//
#include <hip/hip_runtime.h>
#include <hip/hip_bf16.h>

// ---------------------------------------------------------------------------
// Wasserstein similarity for N=256 Gaussians, D=32.
//   out[i,l] = <loc_i, loc_l> + trace( sqrtm( S_i^{1/2} Sigma_l S_i^{1/2} ) )
// All sqrtm's replaced by Newton-Schulz coupled iterations -> pure 32x32 GEMMs
// -> V_WMMA_F32_16X16X4_F32 (2x2 tiles x 8 K-chunks = 32 WMMA per GEMM).
// Iteration 0 is done analytically (Z0 = I), so no GEMM-by-identity and no
// identity writes to LDS.
// ---------------------------------------------------------------------------

typedef __attribute__((ext_vector_type(2))) float v2f;
typedef __attribute__((ext_vector_type(8))) float v8f;

#define NMAT   256
#define DDIM   32
#define PITCH  33            // LDS row pitch (floats); breaks 64-bank aliasing
#define MSZ    (32 * PITCH)  // one 32x32 pitched matrix in LDS
#define WPB    4             // waves per block (128 threads, wave32)
#define ITERS_SQRTM 20       // NS iterations for S_i^{1/2}
#define ITERS_PAIR  16       // NS iterations per pair

__device__ __forceinline__ v8f wmma4(v2f a, v2f b, v8f c) {
  // D = A(16x4,f32) * B(4x16,f32) + C(16x16,f32)
  return __builtin_amdgcn_wmma_f32_16x16x4_f32(
      /*neg_a=*/false, a, /*neg_b=*/false, b,
      /*c_mod=*/(short)0, c, /*reuse_a=*/false, /*reuse_b=*/false);
}

// C[bi*2+bj] = A(32x32, LDS pitched) * B(32x32, LDS pitched)
__device__ __forceinline__ void mm32(const float* A, const float* B, v8f C[4]) {
  const int lane = threadIdx.x & 31;
  const int half = lane >> 4;   // K sub-pair select (ISA A/B f32 layout)
  const int l15  = lane & 15;
  v8f z = {};
  C[0] = z; C[1] = z; C[2] = z; C[3] = z;
#pragma unroll
  for (int kc = 0; kc < 8; ++kc) {
    const int k0 = kc * 4 + half * 2;
    v2f a0, a1, b0, b1;
    a0.x = A[l15 * PITCH + k0];            // A tile rows 0..15
    a0.y = A[l15 * PITCH + k0 + 1];
    a1.x = A[(l15 + 16) * PITCH + k0];     // A tile rows 16..31
    a1.y = A[(l15 + 16) * PITCH + k0 + 1];
    b0.x = B[k0 * PITCH + l15];            // B tile cols 0..15
    b0.y = B[(k0 + 1) * PITCH + l15];
    b1.x = B[k0 * PITCH + l15 + 16];       // B tile cols 16..31
    b1.y = B[(k0 + 1) * PITCH + l15 + 16];
    C[0] = wmma4(a0, b0, C[0]);
    C[1] = wmma4(a0, b1, C[1]);
    C[2] = wmma4(a1, b0, C[2]);
    C[3] = wmma4(a1, b1, C[3]);
  }
}

// Store register C tiles (ISA 16x16 f32 C layout) into pitched LDS matrix.
__device__ __forceinline__ void st32(float* M, const v8f C[4]) {
  const int lane = threadIdx.x & 31;
  const int half = lane >> 4;
  const int l15  = lane & 15;
#pragma unroll
  for (int bi = 0; bi < 2; ++bi)
#pragma unroll
    for (int bj = 0; bj < 2; ++bj)
#pragma unroll
      for (int r = 0; r < 8; ++r)
        M[(bi * 16 + half * 8 + r) * PITCH + bj * 16 + l15] = C[bi * 2 + bj][r];
}

// Load a pitched LDS matrix into register C-tile layout (mirror of st32).
__device__ __forceinline__ void ld32(const float* M, v8f C[4]) {
  const int lane = threadIdx.x & 31;
  const int half = lane >> 4;
  const int l15  = lane & 15;
#pragma unroll
  for (int bi = 0; bi < 2; ++bi)
#pragma unroll
    for (int bj = 0; bj < 2; ++bj)
#pragma unroll
      for (int r = 0; r < 8; ++r)
        C[bi * 2 + bj][r] = M[(bi * 16 + half * 8 + r) * PITCH + bj * 16 + l15];
}

// U = 1.5*I - 0.5*T, done in the C-register layout (diag: l15 == half*8 + r).
__device__ __forceinline__ void ushift(const v8f T[4], v8f U[4]) {
  const int lane = threadIdx.x & 31;
  const int half = lane >> 4;
  const int l15  = lane & 15;
#pragma unroll
  for (int bi = 0; bi < 2; ++bi)
#pragma unroll
    for (int bj = 0; bj < 2; ++bj)
#pragma unroll
      for (int r = 0; r < 8; ++r) {
        float u = -0.5f * T[bi * 2 + bj][r];
        if (bi == bj && l15 == half * 8 + r) u += 1.5f;
        U[bi * 2 + bj][r] = u;
      }
}

// trace of a 32x32 matrix held as C-register tiles; result in all lanes.
__device__ __forceinline__ float trace32(const v8f C[4]) {
  const int lane = threadIdx.x & 31;
  const int half = lane >> 4;
  const int l15  = lane & 15;
  float t = 0.0f;
#pragma unroll
  for (int bi = 0; bi < 2; ++bi)
#pragma unroll
    for (int r = 0; r < 8; ++r)
      if (l15 == half * 8 + r) t += C[bi * 3][r];  // tiles (0,0),(1,1)
#pragma unroll
  for (int off = 16; off >= 1; off >>= 1) t += __shfl_xor(t, off, 32);
  return t;
}

// Coupled Newton-Schulz square root.
// Preconditions: Yb (LDS) holds Y0 = M/||M||_F, Y0r (regs) holds the same in
// C-tile layout. Zb/Ub are scratch (never pre-initialized: Z0 = I is handled
// analytically). On exit Yout (regs) = Y_iters ~= (M/||M||_F)^{1/2}.
__device__ __forceinline__ void ns_sqrt(const v8f Y0r[4], float* Yb, float* Zb,
                                        float* Ub, int iters, v8f Yout[4]) {
  // Iteration 0 (Z0 = I): U0 = 1.5I - 0.5*Y0 ; Z1 = U0 ; Y1 = Y0*U0.
  {
    v8f U[4];
    ushift(Y0r, U);
    st32(Ub, U);
    st32(Zb, U);                // Z1 = U0
    mm32(Yb, Ub, Yout);         // Y1 = Y0*U0
    if (iters == 1) return;
    st32(Yb, Yout);
  }
  for (int it = 1; it < iters; ++it) {
    v8f T[4];
    mm32(Zb, Yb, T);            // T = Z*Y
    v8f U[4];
    ushift(T, U);               // U = 1.5I - 0.5T
    st32(Ub, U);
    mm32(Yb, Ub, Yout);         // Ynew = Y*U
    if (it + 1 == iters) return;  // last iteration: Z not needed
    v8f Zn[4];
    mm32(Ub, Zb, Zn);           // Znew = U*Z
    st32(Yb, Yout);
    st32(Zb, Zn);
  }
}

// ---------------------------------------------------------------------------
// Kernel 1: W[i] = sqrtm(Sigma_i), one wave per matrix.
// ---------------------------------------------------------------------------
__global__ __launch_bounds__(128) void wass_sqrtm(const float* __restrict__ scale,
                                                  float* __restrict__ W) {
  __shared__ float lds[WPB][3 * MSZ];
  const int wave = threadIdx.x >> 5;
  const int lane = threadIdx.x & 31;
  const int i = blockIdx.x * WPB + wave;
  float* Yb = lds[wave];
  float* Zb = lds[wave] + MSZ;
  float* Ub = lds[wave] + 2 * MSZ;

  const float* S = scale + (size_t)i * 1024;
  float f2 = 0.0f;
  for (int idx = lane; idx < 1024; idx += 32) {   // coalesced load + frob^2
    const float v = S[idx];
    Yb[(idx >> 5) * PITCH + (idx & 31)] = v;
    f2 += v * v;
  }
#pragma unroll
  for (int off = 16; off >= 1; off >>= 1) f2 += __shfl_xor(f2, off, 32);
  const float s   = sqrtf(f2);          // ||Sigma||_F >= lambda_max
  const float inv = 1.0f / s;
  for (int idx = lane; idx < 1024; idx += 32)
    Yb[(idx >> 5) * PITCH + (idx & 31)] *= inv;

  v8f Y0r[4];
  ld32(Yb, Y0r);                        // Y0 in C-tile register layout
  v8f Yf[4];
  ns_sqrt(Y0r, Yb, Zb, Ub, ITERS_SQRTM, Yf);

  const float sc = sqrtf(s);            // sqrtm(Sigma) = sqrt(s) * Yf
  const int half = lane >> 4, l15 = lane & 15;
  float* Wo = W + (size_t)i * 1024;     // unpitched row-major in workspace
#pragma unroll
  for (int bi = 0; bi < 2; ++bi)
#pragma unroll
    for (int bj = 0; bj < 2; ++bj)
#pragma unroll
      for (int r = 0; r < 8; ++r)
        Wo[(bi * 16 + half * 8 + r) * 32 + bj * 16 + l15] =
            Yf[bi * 2 + bj][r] * sc;
}

// ---------------------------------------------------------------------------
// Kernel 2: one wave per pair (i,l).
//   cross = W_i * Sigma_l * W_i ;  bures = trace(sqrtm(cross)) via NS
//   out[i*256+l] = <loc_i,loc_l> + bures
// ---------------------------------------------------------------------------
__global__ __launch_bounds__(128) void wass_pair(const float* __restrict__ loc,
                                                 const float* __restrict__ scale,
                                                 const float* __restrict__ W,
                                                 float* __restrict__ out) {
  __shared__ float lds[WPB][3 * MSZ];
  const int wave = threadIdx.x >> 5;
  const int lane = threadIdx.x & 31;
  const int pair = blockIdx.x * WPB + wave;
  const int i = pair >> 8;
  const int l = pair & 255;
  float* Yb = lds[wave];
  float* Zb = lds[wave] + MSZ;
  float* Ub = lds[wave] + 2 * MSZ;

  // Euclidean part: D=32 == wave width, one element per lane.
  float e = loc[i * DDIM + lane] * loc[l * DDIM + lane];
#pragma unroll
  for (int off = 16; off >= 1; off >>= 1) e += __shfl_xor(e, off, 32);

  // Stage W_i -> Zb, Sigma_l -> Yb (pitched LDS).
  const float* Wi = W + (size_t)i * 1024;
  const float* Sl = scale + (size_t)l * 1024;
  for (int idx = lane; idx < 1024; idx += 32) {
    const int r = idx >> 5, c = idx & 31;
    Zb[r * PITCH + c] = Wi[idx];
    Yb[r * PITCH + c] = Sl[idx];
  }

  // cross = (W * Sigma) * W
  v8f P[4];
  mm32(Zb, Yb, P);              // P = W * Sigma
  st32(Ub, P);
  v8f Cx[4];
  mm32(Ub, Zb, Cx);             // Cx = P * W   (in C-register layout)

  // Frobenius-normalize cross.
  float f2 = 0.0f;
#pragma unroll
  for (int q = 0; q < 4; ++q)
#pragma unroll
    for (int r = 0; r < 8; ++r) f2 += Cx[q][r] * Cx[q][r];
#pragma unroll
  for (int off = 16; off >= 1; off >>= 1) f2 += __shfl_xor(f2, off, 32);
  const float s   = sqrtf(f2);
  const float inv = 1.0f / s;
#pragma unroll
  for (int q = 0; q < 4; ++q)
#pragma unroll
    for (int r = 0; r < 8; ++r) Cx[q][r] *= inv;

  st32(Yb, Cx);                 // Y0 = cross / ||cross||_F (LDS copy)

  v8f Yf[4];
  ns_sqrt(Cx, Yb, Zb, Ub, ITERS_PAIR, Yf);   // Y0 regs passed directly

  const float bures = sqrtf(s) * trace32(Yf);
  if (lane == 0) out[pair] = e + bures;
}

// ---------------------------------------------------------------------------
extern "C" void kernel_launch(void* const* d_in, const int* in_sizes, int n_in,
                              void* d_out, int out_size, void* d_ws, size_t ws_size,
                              hipStream_t stream) {
  const float* loc   = (const float*)d_in[0];   // [256,32]  f32
  const float* scale = (const float*)d_in[1];   // [256,32,32] f32
  float* out = (float*)d_out;                   // [256,256] f32
  float* W   = (float*)d_ws;                    // 256*1024 f32 = 1 MB scratch

  (void)in_sizes; (void)n_in; (void)out_size; (void)ws_size;

  // Stage 1: 256 matrix square roots, one wave each (4 waves/block).
  wass_sqrtm<<<NMAT / WPB, 32 * WPB, 0, stream>>>(scale, W);
  // Stage 2: 65536 pairs, one wave each.
  wass_pair<<<(NMAT * NMAT) / WPB, 32 * WPB, 0, stream>>>(loc, scale, W, out);
}